// FeatureLevelAttention_14937896255681
// MI455X (gfx1250) — compile-verified
//
#include <hip/hip_runtime.h>
#include <math.h>

typedef float v2f __attribute__((ext_vector_type(2)));
typedef float v8f __attribute__((ext_vector_type(8)));

#define BATCH 16
#define CIN_  3
#define HW    16384
#define CCH   256
#define NH    8
#define HD    32
#define SCALE_F 0.17677669529663687f
#define EPS_F  1e-5f

#define OUT_MAIN (BATCH*CCH*HW)       /* 67108864 */
#define S_ELEMS  (BATCH*NH*HD*HD)     /* 131072   */

#define PTILE 32
#define NCH   64                      /* pixel chunks per batch image */
#define X1S   260                     /* x1/stage buffer row stride (words), ==4 mod 64 */
#define QKS   36                      /* qk buffer row stride */
#define ATS   36                      /* attn LDS row stride */

// One 16x16 f32 WMMA K-loop with two B-column tiles sharing the A operand.
// Lane mapping (per CDNA5 ISA): m = lane&15 (A row / B col), hf = lane>>4 (K pair).
__device__ __forceinline__ void mma_loop2(const float* aP, const float* bP0, const float* bP1,
                                          int kSteps, v8f& acc0, v8f& acc1) {
#pragma unroll 4
  for (int t = 0; t < kSteps; ++t) {
    v2f a  = *(const v2f*)aP;
    v2f b0 = *(const v2f*)bP0;
    v2f b1 = *(const v2f*)bP1;
    acc0 = __builtin_amdgcn_wmma_f32_16x16x4_f32(false, a, false, b0, (short)0, acc0, false, false);
    acc1 = __builtin_amdgcn_wmma_f32_16x16x4_f32(false, a, false, b1, (short)0, acc1, false, false);
    aP += 4; bP0 += 4; bP1 += 4;
  }
}

// Recompute x1[:,p0..p0+31] = Wi @ x + bi into LDS (pixel-major, stride X1S).
__device__ __forceinline__ void fill_x1(float* sX1, const float* __restrict__ x,
                                        const float* __restrict__ Wi, const float* __restrict__ bi,
                                        int b, int p0, int t) {
  int p  = t & 31;          // pixel
  int cg = t >> 5;          // channel group 0..7 (32 channels each)
  const float* xp = x + (size_t)b * CIN_ * HW + p0 + p;
  float x0 = xp[0], x1v = xp[HW], x2v = xp[2 * HW];
#pragma unroll
  for (int j = 0; j < 32; ++j) {
    int c = cg * 32 + j;
    sX1[p * X1S + c] = Wi[c * 3 + 0] * x0 + Wi[c * 3 + 1] * x1v + Wi[c * 3 + 2] * x2v + bi[c];
  }
}

__global__ void zeroS_kernel(float* __restrict__ S) {
  int i = blockIdx.x * 256 + threadIdx.x;
  if (i < S_ELEMS) S[i] = 0.f;
}

// Pass 1: per (batch, pixel-chunk): x1 tile -> q,k rows of Wqkv -> S += q k^T (per head).
__global__ __launch_bounds__(256) void qk_s_kernel(const float* __restrict__ x,
                                                   const float* __restrict__ Wi,
                                                   const float* __restrict__ bi,
                                                   const float* __restrict__ Wqkv,
                                                   float* __restrict__ S) {
  extern __shared__ float smem[];
  float* sX1 = smem;                 // 32 x 260
  float* sQK = smem + 32 * X1S;      // 512 rows (q,k per head) x 36

  const int t    = threadIdx.x;
  const int wave = t >> 5;
  const int lane = t & 31;
  const int m    = lane & 15;
  const int hf   = lane >> 4;
  const int b     = blockIdx.x / NCH;
  const int chunk = blockIdx.x % NCH;
  const int pBase = chunk * (HW / NCH);   // 256 pixels per chunk

  v8f sAcc[4];
#pragma unroll
  for (int i = 0; i < 4; ++i) sAcc[i] = (v8f){0,0,0,0,0,0,0,0};

  for (int it = 0; it < (HW / NCH) / PTILE; ++it) {
    int p0 = pBase + it * PTILE;
    __syncthreads();
    fill_x1(sX1, x, Wi, bi, b, p0, t);
    __syncthreads();

    // q,k GEMM: 32 row-tiles (512 rows) x 2 col-tiles; 4 row-tiles per wave.
#pragma unroll
    for (int j = 0; j < 4; ++j) {
      int rt  = wave + j * 8;          // 0..31
      int n   = rt >> 2;               // head
      int sub = rt & 3;                // 0-1: q rows, 2-3: k rows
      int gRow = n * 96 + sub * 16 + m;        // Wqkv row (q:n*96+d, k:n*96+32+d)
      const float* aP  = Wqkv + (size_t)gRow * CCH + 2 * hf;
      const float* bP0 = sX1 + (0  + m) * X1S + 2 * hf;
      const float* bP1 = sX1 + (16 + m) * X1S + 2 * hf;
      v8f a0 = (v8f){0,0,0,0,0,0,0,0}, a1 = (v8f){0,0,0,0,0,0,0,0};
      mma_loop2(aP, bP0, bP1, CCH / 4, a0, a1);
      int rowL = n * 64 + sub * 16;
#pragma unroll
      for (int r = 0; r < 8; ++r) {
        int me = r + 8 * hf;
        sQK[(rowL + me) * QKS + 0  + m] = a0[r];
        sQK[(rowL + me) * QKS + 16 + m] = a1[r];
      }
    }
    __syncthreads();

    // S accumulation: wave == head; K = 32 pixels.
    {
      int n = wave;
#pragma unroll
      for (int di = 0; di < 2; ++di) {
        const float* aP  = sQK + (n * 64 + di * 16 + m) * QKS + 2 * hf;       // q rows
        const float* bP0 = sQK + (n * 64 + 32 + 0  + m) * QKS + 2 * hf;       // k rows e 0..15
        const float* bP1 = sQK + (n * 64 + 32 + 16 + m) * QKS + 2 * hf;       // k rows e 16..31
        mma_loop2(aP, bP0, bP1, PTILE / 4, sAcc[di * 2 + 0], sAcc[di * 2 + 1]);
      }
    }
  }

  // Fold partial S into global accumulator.
  {
    int n = wave;
#pragma unroll
    for (int tl = 0; tl < 4; ++tl) {
      int di = tl >> 1, ei = tl & 1;
#pragma unroll
      for (int r = 0; r < 8; ++r) {
        int d = di * 16 + r + 8 * hf;
        int e = ei * 16 + m;
        atomicAdd(&S[(((size_t)b * NH + n) * HD + d) * HD + e], sAcc[tl][r]);
      }
    }
  }
}

__global__ void attn_kernel(const float* __restrict__ S, float* __restrict__ attnWs,
                            float* __restrict__ outTail) {
  int i = blockIdx.x * 256 + threadIdx.x;
  if (i < S_ELEMS) {
    float v = S[i] * SCALE_F;
    float a = 1.f / (1.f + __expf(-v));
    attnWs[i] = a;
    outTail[i] = a;
  }
}

// Pass 2: x1 -> v -> attn@v -> Wo -> Wf -> BN -> SiLU -> +x1 -> out
__global__ __launch_bounds__(256) void out_kernel(const float* __restrict__ x,
                                                  const float* __restrict__ Wi,
                                                  const float* __restrict__ bi,
                                                  const float* __restrict__ Wqkv,
                                                  const float* __restrict__ Wo,
                                                  const float* __restrict__ bo,
                                                  const float* __restrict__ Wf,
                                                  const float* __restrict__ bf,
                                                  const float* __restrict__ gamma,
                                                  const float* __restrict__ beta,
                                                  const float* __restrict__ rmean,
                                                  const float* __restrict__ rvar,
                                                  const float* __restrict__ attn,
                                                  float* __restrict__ out) {
  extern __shared__ float smem[];
  float* sX1 = smem;                    // 32 x 260
  float* sA  = smem + 32 * X1S;         // stage buffer A
  float* sB  = smem + 2 * 32 * X1S;     // stage buffer B
  float* sAt = smem + 3 * 32 * X1S;     // attn (256 rows x 36)

  const int t    = threadIdx.x;
  const int wave = t >> 5;
  const int lane = t & 31;
  const int m    = lane & 15;
  const int hf   = lane >> 4;
  const int b     = blockIdx.x / NCH;
  const int chunk = blockIdx.x % NCH;
  const int pBase = chunk * (HW / NCH);

  // Stage this batch's attention matrices into LDS (padded rows).
  for (int i = t; i < NH * HD * HD; i += 256) {
    int row = i >> 5;   // n*32 + d
    int e   = i & 31;
    sAt[row * ATS + e] = attn[(size_t)b * NH * HD * HD + i];
  }

  for (int it = 0; it < (HW / NCH) / PTILE; ++it) {
    int p0 = pBase + it * PTILE;
    __syncthreads();
    fill_x1(sX1, x, Wi, bi, b, p0, t);
    __syncthreads();

    // Stage V: v = Wv @ x1 (v rows of Wqkv: n*96+64+e) -> sA (pixel-major, c = n*32+e)
#pragma unroll
    for (int j = 0; j < 2; ++j) {
      int rt = wave + j * 8;            // 0..15
      int n  = rt >> 1;
      int e0 = (rt & 1) * 16;
      const float* aP  = Wqkv + (size_t)(n * 96 + 64 + e0 + m) * CCH + 2 * hf;
      const float* bP0 = sX1 + (0  + m) * X1S + 2 * hf;
      const float* bP1 = sX1 + (16 + m) * X1S + 2 * hf;
      v8f a0 = (v8f){0,0,0,0,0,0,0,0}, a1 = (v8f){0,0,0,0,0,0,0,0};
      mma_loop2(aP, bP0, bP1, CCH / 4, a0, a1);
#pragma unroll
      for (int r = 0; r < 8; ++r) {
        int c = rt * 16 + r + 8 * hf;
        sA[(0  + m) * X1S + c] = a0[r];
        sA[(16 + m) * X1S + c] = a1[r];
      }
    }
    __syncthreads();

    // Stage attn@v (K = 32 per head) -> sB
#pragma unroll
    for (int j = 0; j < 2; ++j) {
      int rt = wave + j * 8;
      int n  = rt >> 1;
      int d0 = (rt & 1) * 16;
      const float* aP  = sAt + (n * HD + d0 + m) * ATS + 2 * hf;
      const float* bP0 = sA + (0  + m) * X1S + n * HD + 2 * hf;
      const float* bP1 = sA + (16 + m) * X1S + n * HD + 2 * hf;
      v8f a0 = (v8f){0,0,0,0,0,0,0,0}, a1 = (v8f){0,0,0,0,0,0,0,0};
      mma_loop2(aP, bP0, bP1, HD / 4, a0, a1);
#pragma unroll
      for (int r = 0; r < 8; ++r) {
        int c = n * HD + d0 + r + 8 * hf;
        sB[(0  + m) * X1S + c] = a0[r];
        sB[(16 + m) * X1S + c] = a1[r];
      }
    }
    __syncthreads();

    // Stage Wo @ A + bo -> sA
#pragma unroll
    for (int j = 0; j < 2; ++j) {
      int rt = wave + j * 8;
      const float* aP  = Wo + (size_t)(rt * 16 + m) * CCH + 2 * hf;
      const float* bP0 = sB + (0  + m) * X1S + 2 * hf;
      const float* bP1 = sB + (16 + m) * X1S + 2 * hf;
      v8f a0 = (v8f){0,0,0,0,0,0,0,0}, a1 = (v8f){0,0,0,0,0,0,0,0};
      mma_loop2(aP, bP0, bP1, CCH / 4, a0, a1);
#pragma unroll
      for (int r = 0; r < 8; ++r) {
        int o = rt * 16 + r + 8 * hf;
        float bb = bo[o];
        sA[(0  + m) * X1S + o] = a0[r] + bb;
        sA[(16 + m) * X1S + o] = a1[r] + bb;
      }
    }
    __syncthreads();

    // Stage Wf @ t1 + bf, then BN + SiLU + residual, store to global out.
#pragma unroll
    for (int j = 0; j < 2; ++j) {
      int rt = wave + j * 8;
      const float* aP  = Wf + (size_t)(rt * 16 + m) * CCH + 2 * hf;
      const float* bP0 = sA + (0  + m) * X1S + 2 * hf;
      const float* bP1 = sA + (16 + m) * X1S + 2 * hf;
      v8f a0 = (v8f){0,0,0,0,0,0,0,0}, a1 = (v8f){0,0,0,0,0,0,0,0};
      mma_loop2(aP, bP0, bP1, CCH / 4, a0, a1);
#pragma unroll
      for (int r = 0; r < 8; ++r) {
        int o = rt * 16 + r + 8 * hf;
        float inv = gamma[o] * rsqrtf(rvar[o] + EPS_F);
        float mu = rmean[o], bt = beta[o], bfo = bf[o];
        float* orow = out + ((size_t)b * CCH + o) * HW + p0;
#pragma unroll
        for (int ct = 0; ct < 2; ++ct) {
          float v = (ct ? a1[r] : a0[r]) + bfo;
          v = (v - mu) * inv + bt;
          v = v / (1.f + __expf(-v));      // SiLU
          int p = ct * 16 + m;
          v += sX1[p * X1S + o];           // residual x1
          orow[p] = v;
        }
      }
    }
  }
}

extern "C" void kernel_launch(void* const* d_in, const int* in_sizes, int n_in,
                              void* d_out, int out_size, void* d_ws, size_t ws_size,
                              hipStream_t stream) {
  (void)in_sizes; (void)n_in; (void)out_size; (void)ws_size;
  const float* x     = (const float*)d_in[0];
  const float* Wi    = (const float*)d_in[1];
  const float* bi    = (const float*)d_in[2];
  const float* Wqkv  = (const float*)d_in[3];
  const float* Wo    = (const float*)d_in[4];
  const float* bo    = (const float*)d_in[5];
  const float* Wf    = (const float*)d_in[6];
  const float* bf    = (const float*)d_in[7];
  const float* gamma = (const float*)d_in[8];
  const float* beta  = (const float*)d_in[9];
  const float* rmean = (const float*)d_in[10];
  const float* rvar  = (const float*)d_in[11];
  float* out  = (float*)d_out;
  float* S    = (float*)d_ws;        // 131072 floats
  float* attnW = S + S_ELEMS;        // 131072 floats

  const size_t smemS = (size_t)(32 * X1S + 512 * QKS) * sizeof(float);              // ~107 KB
  const size_t smemO = (size_t)(3 * 32 * X1S + NH * HD * ATS) * sizeof(float);      // ~137 KB

  zeroS_kernel<<<(S_ELEMS + 255) / 256, 256, 0, stream>>>(S);
  qk_s_kernel<<<BATCH * NCH, 256, smemS, stream>>>(x, Wi, bi, Wqkv, S);
  attn_kernel<<<(S_ELEMS + 255) / 256, 256, 0, stream>>>(S, attnW, out + OUT_MAIN);
  out_kernel<<<BATCH * NCH, 256, smemO, stream>>>(x, Wi, bi, Wqkv, Wo, bo, Wf, bf,
                                                  gamma, beta, rmean, rvar, attnW, out);
}